// HungarianMatcher_54030688584050
// MI455X (gfx1250) — compile-verified
//
#include <hip/hip_runtime.h>
#include <cmath>

// ---------------------------------------------------------------------------
// HungarianMatcher cost matrix on MI455X (gfx1250)
//
// Factorization: polynomial L2 cost becomes a bilinear form over 16 moment
// features + 3 class one-hot dims  =>  2048x480 GEMM with K=20 computed with
// V_WMMA_F32_16X16X4_F32 (full fp32 precision; lambda^6 moments would
// overflow f16).  |.| lower/upper terms are added in a VALU epilogue.
// ---------------------------------------------------------------------------

constexpr int BS = 16, NQ = 128, NCLS = 3, T = 480, P = 72;
constexpr int BN = BS * NQ;                 // 2048 queries
constexpr int TGT_STRIDE = 5 + 2 * P;       // 149 floats per target row
constexpr int KF = 32;                      // padded feature stride (20 used)

typedef __attribute__((ext_vector_type(2))) float v2f;
typedef __attribute__((ext_vector_type(8))) float v8f;

// ---------------------------------------------------------------------------
// Kernel Q: per-query feature rows  Aq[BN][KF]
//   f0 = 1, f1..4 = a0..a3, f5..8 = b0..b3, f9..15 = c_m (m=0..6),
//   f16..18 = -softmax(logits), f19..31 = 0
// ---------------------------------------------------------------------------
__global__ void hm_query_features(const float* __restrict__ logits,
                                  const float* __restrict__ curves,
                                  float* __restrict__ Aq) {
    int i = blockIdx.x * blockDim.x + threadIdx.x;
    if (i >= BN) return;
    const float* ob = curves + i * 8;
    float ob0 = ob[0], ob1 = ob[1], ob2 = ob[2], ob3 = ob[3];
    float ob4 = ob[4], ob5 = ob[5], ob6 = ob[6], ob7 = ob[7];
    float a0 = ob2, a3 = ob4, a2 = ob5, a1 = ob3 - ob4 - ob5 - ob2;
    float b0 = ob0, b3 = ob6, b2 = ob7, b1 = ob1 - ob6 - ob7 - ob0;

    const float* lg = logits + i * NCLS;
    float l0 = lg[0], l1 = lg[1], l2 = lg[2];
    float mx = fmaxf(l0, fmaxf(l1, l2));
    float e0 = __expf(l0 - mx), e1 = __expf(l1 - mx), e2 = __expf(l2 - mx);
    float inv = 1.0f / (e0 + e1 + e2);

    float f[KF];
    f[0] = 1.0f;
    f[1] = a0; f[2] = a1; f[3] = a2; f[4] = a3;
    f[5] = b0; f[6] = b1; f[7] = b2; f[8] = b3;
    f[9]  = a0 * a0 + b0 * b0;
    f[10] = 2.0f * (a0 * a1 + b0 * b1);
    f[11] = 2.0f * (a0 * a2 + b0 * b2) + a1 * a1 + b1 * b1;
    f[12] = 2.0f * (a0 * a3 + a1 * a2 + b0 * b3 + b1 * b2);
    f[13] = 2.0f * (a1 * a3 + b1 * b3) + a2 * a2 + b2 * b2;
    f[14] = 2.0f * (a2 * a3 + b2 * b3);
    f[15] = a3 * a3 + b3 * b3;
    f[16] = -e0 * inv; f[17] = -e1 * inv; f[18] = -e2 * inv;
#pragma unroll
    for (int k = 19; k < KF; ++k) f[k] = 0.0f;
#pragma unroll
    for (int k = 0; k < KF; ++k) Aq[i * KF + k] = f[k];
}

// ---------------------------------------------------------------------------
// Kernel T: per-target raw moments (unscaled)  Fraw[T][KF], per[T]
// ---------------------------------------------------------------------------
__global__ void hm_target_raw(const float* __restrict__ targets,
                              float* __restrict__ Fraw,
                              float* __restrict__ per) {
    int t = blockIdx.x * blockDim.x + threadIdx.x;
    if (t >= T) return;
    const float* row = targets + t * TGT_STRIDE;
    float ly = row[1], uy = row[2], lx = row[3], ux = row[4];
    float dx = ux - lx, dy = uy - ly;
    float inv_n2 = 1.0f / (dx * dx + dy * dy);

    float mu0 = 0, mu1 = 0, mu2 = 0, mu3 = 0, mu4 = 0, mu5 = 0, mu6 = 0;
    float Mx0 = 0, Mx1 = 0, Mx2 = 0, Mx3 = 0;
    float Ny0 = 0, Ny1 = 0, Ny2 = 0, Ny3 = 0;
    float S = 0;

    for (int p = 0; p < P; ++p) {
        float tx = row[5 + p];
        float ty = row[5 + P + p];
        if (tx >= 0.0f) {
            float L  = ((tx - lx) * dx + (ty - ly) * dy) * inv_n2;
            float L2 = L * L, L3 = L2 * L;
            float L4 = L2 * L2, L5 = L4 * L, L6 = L3 * L3;
            mu0 += 1.0f; mu1 += L;  mu2 += L2; mu3 += L3;
            mu4 += L4;   mu5 += L5; mu6 += L6;
            Mx0 += tx; Mx1 += tx * L; Mx2 += tx * L2; Mx3 += tx * L3;
            Ny0 += ty; Ny1 += ty * L; Ny2 += ty * L2; Ny3 += ty * L3;
            S += tx * tx + ty * ty;
        }
    }
    float* F = Fraw + t * KF;
    F[0] = S;
    F[1] = -2.0f * Mx0; F[2] = -2.0f * Mx1; F[3] = -2.0f * Mx2; F[4] = -2.0f * Mx3;
    F[5] = -2.0f * Ny0; F[6] = -2.0f * Ny1; F[7] = -2.0f * Ny2; F[8] = -2.0f * Ny3;
    F[9] = mu0; F[10] = mu1; F[11] = mu2; F[12] = mu3;
    F[13] = mu4; F[14] = mu5; F[15] = mu6;
    per[t] = mu0;
}

// ---------------------------------------------------------------------------
// Kernel B: min-reduce per[], then Ft[t] = Fraw[t]*(0.1*sqrt(min/per)) plus
// class one-hot (slots 16..18) and zero padding. Single block of 512.
//   w[t]/max(w) = sqrt(min_per / per[t])  (total cancels)
// ---------------------------------------------------------------------------
__global__ void hm_target_finalize(const float* __restrict__ targets,
                                   const float* __restrict__ Fraw,
                                   const float* __restrict__ per,
                                   float* __restrict__ Ft) {
    __shared__ float sm[512];
    int tid = threadIdx.x;
    sm[tid] = (tid < T) ? per[tid] : 3.0e38f;
    __syncthreads();
    for (int s = 256; s > 0; s >>= 1) {
        if (tid < s) sm[tid] = fminf(sm[tid], sm[tid + s]);
        __syncthreads();
    }
    float min_per = sm[0];
    if (tid < T) {
        int t = tid;
        float scale = 0.1f * sqrtf(min_per / per[t]);  // CURVES_W * w / 10
        int id = (int)targets[t * TGT_STRIDE];
        const float* src = Fraw + t * KF;
        float* dst = Ft + t * KF;
#pragma unroll
        for (int k = 0; k < 16; ++k) dst[k] = src[k] * scale;
        dst[16] = (id == 0) ? 1.0f : 0.0f;
        dst[17] = (id == 1) ? 1.0f : 0.0f;
        dst[18] = (id == 2) ? 1.0f : 0.0f;
#pragma unroll
        for (int k = 19; k < KF; ++k) dst[k] = 0.0f;
    }
}

// ---------------------------------------------------------------------------
// Kernel G: 2048x480 GEMM (K=20) via V_WMMA_F32_16X16X4_F32, one wave32 per
// 16x16 tile, + abs-cost epilogue.
//   A frag (16x4 f32): lane L holds m=L%16, components = K 2*(L/16)+{0,1}
//   B frag (4x16 f32): assumed symmetric — lane L holds n=L%16, same K split
//   D frag (16x16 f32): lane L holds n=L%16, acc[r] = row r+8*(L/16)
// ---------------------------------------------------------------------------
__global__ void __launch_bounds__(128)
hm_gemm_epilogue(const float* __restrict__ Aq,
                 const float* __restrict__ Ft,
                 const float* __restrict__ targets,
                 const float* __restrict__ curves,
                 float* __restrict__ out) {
    int lane = threadIdx.x & 31;
    int wave = threadIdx.x >> 5;
    int tileN = blockIdx.x;               // 0..29  (480 / 16)
    int tileM = blockIdx.y * 4 + wave;    // 0..127 (2048 / 16)
    int half = lane >> 4;
    int ln = lane & 15;

    const float* arow = Aq + (tileM * 16 + ln) * KF + 2 * half;
    const float* brow = Ft + (tileN * 16 + ln) * KF + 2 * half;

    v8f acc = {};
#pragma unroll
    for (int s = 0; s < 5; ++s) {        // K = 20 = 5 * 4
        v2f a; a.x = arow[4 * s]; a.y = arow[4 * s + 1];
        v2f b; b.x = brow[4 * s]; b.y = brow[4 * s + 1];
        acc = __builtin_amdgcn_wmma_f32_16x16x4_f32(
            /*neg_a=*/false, a, /*neg_b=*/false, b,
            /*c_mod=*/(short)0, acc, /*reuse_a=*/false, /*reuse_b=*/false);
    }

    // Epilogue: lane owns column t = tileN*16 + ln of the D tile.
    int t = tileN * 16 + ln;
    const float* trow = targets + t * TGT_STRIDE;
    float ly = trow[1], uy = trow[2], lx = trow[3], ux = trow[4];
#pragma unroll
    for (int r = 0; r < 8; ++r) {
        int i = tileM * 16 + r + 8 * half;
        const float* ob = curves + i * 8;
        float add = 0.5f * (fabsf(ob[0] - ly) + fabsf(ob[2] - lx))   // lower
                  + 0.5f * (fabsf(ob[1] - uy) + fabsf(ob[3] - ux));  // upper
        out[i * T + t] = acc[r] + add;
    }
}

// ---------------------------------------------------------------------------
extern "C" void kernel_launch(void* const* d_in, const int* in_sizes, int n_in,
                              void* d_out, int out_size, void* d_ws, size_t ws_size,
                              hipStream_t stream) {
    const float* logits  = (const float*)d_in[0];   // (16,128,3)
    const float* curves  = (const float*)d_in[1];   // (16,128,8)
    const float* targets = (const float*)d_in[2];   // (480,149)
    float* out = (float*)d_out;                     // (16,128,480)

    float* ws   = (float*)d_ws;
    float* Aq   = ws;                    // BN*KF  = 65536 floats
    float* Fraw = Aq + BN * KF;          // T*KF   = 15360 floats
    float* Ft   = Fraw + T * KF;         // T*KF   = 15360 floats
    float* per  = Ft + T * KF;           // T      = 480 floats

    hm_query_features<<<(BN + 127) / 128, 128, 0, stream>>>(logits, curves, Aq);
    hm_target_raw<<<(T + 127) / 128, 128, 0, stream>>>(targets, Fraw, per);
    hm_target_finalize<<<1, 512, 0, stream>>>(targets, Fraw, per, Ft);
    hm_gemm_epilogue<<<dim3(30, 32), 128, 0, stream>>>(Aq, Ft, targets, curves, out);
}